// Correlation_38929583571346
// MI455X (gfx1250) — compile-verified
//
#include <hip/hip_runtime.h>

typedef __attribute__((ext_vector_type(2))) float v2f;
typedef __attribute__((ext_vector_type(8))) float v8f;
typedef __attribute__((ext_vector_type(4))) int   v4i;

#define PADSZ   20
#define NDISP   21          // displacements per axis (441 total)
#define BATCH   8
#define CHANS   256
#define HEIGHT  96
#define WIDTH   128
#define GH      8           // same-parity h rows per block
#define NSLOT   (GH + 20)   // 28 input2 rows feed this strip
#define BSTR    72          // B row stride (floats) -> conflict-free half-wave K split
#define PSTR    68          // P row stride (floats)

// ---- CDNA5 async global->LDS path (ASYNCcnt), with safe fallback -----------
#if defined(__has_builtin)
#if __has_builtin(__builtin_amdgcn_global_load_async_to_lds_b128)
#define HAVE_ASYNC_LDS 1
#endif
#endif

#ifdef HAVE_ASYNC_LDS
#define AS_GLOBAL __attribute__((address_space(1)))
#define AS_LDS    __attribute__((address_space(3)))
static __device__ __forceinline__ void async_copy_b128(const float* g, float* l) {
    __builtin_amdgcn_global_load_async_to_lds_b128(
        (AS_GLOBAL v4i*)(g), (AS_LDS v4i*)(l), /*offset=*/0, /*cpol=*/0);
}
#endif

__global__ __launch_bounds__(256)
void corr_wmma_kernel(const float* __restrict__ in1,
                      const float* __restrict__ in2,
                      float* __restrict__ out)
{
    // LDS: A 128 KB (resident) + B 2x73.7 KB (double buffer) + P 34.8 KB = 313 KB
    __shared__ __align__(16) float Alds[GH][CHANS][16];
    __shared__ __align__(16) float Blds[2][CHANS][BSTR];
    __shared__ __align__(16) float Plds[GH][16][PSTR];

    const int wt    = blockIdx.x;            // 0..7  w-tile
    const int strip = blockIdx.y;            // 0..11 (parity + group)
    const int b     = blockIdx.z;            // 0..7
    const int par   = strip & 1;
    const int grp   = strip >> 1;            // 0..5
    const int hbase = par + (grp << 4);      // h_g = hbase + 2g, g=0..7
    const int w0    = wt * 16;
    const int xbase = w0 - PADSZ;

    const int tid  = threadIdx.x;            // 0..255
    const int lane = tid & 31;
    const int wave = tid >> 5;               // 0..7 == this wave's g
    const int i16  = lane & 15;
    const int koff = (lane >> 4) << 1;       // 0 (lanes 0-15) / 2 (lanes 16-31)
    const float invC = 1.0f / (float)CHANS;
    const size_t planeHW = (size_t)HEIGHT * WIDTH;

    // ---- stage all 8 A tiles once: Alds[g][c][i] = in1[b, c, hbase+2g, w0+i]
    {
        const float* p1 = in1 + (size_t)b * CHANS * planeHW;
        for (int idx = tid; idx < GH * CHANS * 4; idx += 256) {
            int g   = idx >> 10;             // 1024 float4 per tile
            int rem = idx & 1023;
            int c   = rem >> 2;
            int j4  = (rem & 3) << 2;
            float4 v = *(const float4*)(p1 + (size_t)c * planeHW
                                           + (size_t)(hbase + 2 * g) * WIDTH + w0 + j4);
            *(float4*)&Alds[g][c][j4] = v;
        }
    }

    // ---- B staging geometry (256 threads: 16 threads/row, 16 rows/sweep)
    const int  bc0  = tid >> 4;              // 0..15 channel base
    const int  bj4  = (tid & 15) << 2;       // 0..60 column (float4)
    const int  bx0  = xbase + bj4;
    const bool bxok = (bx0 >= 0) && (bx0 + 3 < WIDTH);  // 4-aligned band => exact
    const float* p2base = in2 + (size_t)b * CHANS * planeHW + bx0;

    // OOB band columns are fixed for the whole block: zero them once in both
    // buffers; masked staging below never touches them again.
    if (!bxok) {
        const float4 z = make_float4(0.f, 0.f, 0.f, 0.f);
        #pragma unroll
        for (int r = 0; r < 16; ++r) {
            *(float4*)&Blds[0][bc0 + (r << 4)][bj4] = z;
            *(float4*)&Blds[1][bc0 + (r << 4)][bj4] = z;
        }
    }

#ifndef HAVE_ASYNC_LDS
    float4 breg[16];
#endif

    // ---- prologue: stage slot 0 into buffer 0
    {
        int y = hbase - PADSZ;
        if (y >= 0 && y < HEIGHT) {
            const float* p2 = p2base + (size_t)y * WIDTH;
#ifdef HAVE_ASYNC_LDS
            if (bxok) {
                #pragma unroll
                for (int r = 0; r < 16; ++r) {
                    int c = bc0 + (r << 4);
                    async_copy_b128(p2 + (size_t)c * planeHW, &Blds[0][c][bj4]);
                }
            }
#else
            #pragma unroll
            for (int r = 0; r < 16; ++r) {
                int c = bc0 + (r << 4);
                breg[r] = bxok ? *(const float4*)(p2 + (size_t)c * planeHW)
                               : make_float4(0.f, 0.f, 0.f, 0.f);
            }
            #pragma unroll
            for (int r = 0; r < 16; ++r)
                if (bxok) *(float4*)&Blds[0][bc0 + (r << 4)][bj4] = breg[r];
#endif
        }
    }
#ifdef HAVE_ASYNC_LDS
    asm volatile("s_wait_asynccnt 0" ::: "memory");
#endif
    __syncthreads();

    for (int s = 0; s < NSLOT; ++s) {
        const int  buf = s & 1;
        const int  y   = hbase - PADSZ + 2 * s;
        const bool yok = (y >= 0) && (y < HEIGHT);
        const int  yn  = y + 2;
        const bool nok = (s + 1 < NSLOT) && (yn >= 0) && (yn < HEIGHT);

        // issue staging for next slot into the alternate buffer (overlaps GEMM)
        if (nok) {
            const float* p2 = p2base + (size_t)yn * WIDTH;
#ifdef HAVE_ASYNC_LDS
            if (bxok) {
                #pragma unroll
                for (int r = 0; r < 16; ++r) {
                    int c = bc0 + (r << 4);
                    async_copy_b128(p2 + (size_t)c * planeHW, &Blds[buf ^ 1][c][bj4]);
                }
            }
#else
            #pragma unroll
            for (int r = 0; r < 16; ++r) {
                int c = bc0 + (r << 4);
                breg[r] = bxok ? *(const float4*)(p2 + (size_t)c * planeHW)
                               : make_float4(0.f, 0.f, 0.f, 0.f);
            }
#endif
        }

        // this wave's (h, dy) pair for slot s
        const int g   = wave;
        const int dyi = s - g;
        if (dyi >= 0 && dyi < NDISP) {               // wave-uniform branch
            const int h = hbase + 2 * g;
            float* outp = out + (((size_t)b * (NDISP * NDISP) + (size_t)dyi * NDISP)
                                 * HEIGHT + h) * WIDTH + w0;
            if (yok) {
                v8f acc0 = {0.f,0.f,0.f,0.f,0.f,0.f,0.f,0.f};
                v8f acc1 = acc0, acc2 = acc0, acc3 = acc0;

                // banded GEMM: P[16,64] = A_g[16,256] x B_y[256,64]
                // A pair loaded once per k-step, shared by 4 WMMAs
                for (int k = 0; k < CHANS; k += 4) {
                    v2f a, b0, b1, b2, b3;
                    a[0]  = Alds[g][k + koff    ][i16];
                    a[1]  = Alds[g][k + koff + 1][i16];
                    b0[0] = Blds[buf][k + koff    ][ 0 + i16];
                    b0[1] = Blds[buf][k + koff + 1][ 0 + i16];
                    b1[0] = Blds[buf][k + koff    ][16 + i16];
                    b1[1] = Blds[buf][k + koff + 1][16 + i16];
                    b2[0] = Blds[buf][k + koff    ][32 + i16];
                    b2[1] = Blds[buf][k + koff + 1][32 + i16];
                    b3[0] = Blds[buf][k + koff    ][48 + i16];
                    b3[1] = Blds[buf][k + koff + 1][48 + i16];
                    acc0 = __builtin_amdgcn_wmma_f32_16x16x4_f32(false, a, false, b0,
                                                                 (short)0, acc0, false, false);
                    acc1 = __builtin_amdgcn_wmma_f32_16x16x4_f32(false, a, false, b1,
                                                                 (short)0, acc1, false, false);
                    acc2 = __builtin_amdgcn_wmma_f32_16x16x4_f32(false, a, false, b2,
                                                                 (short)0, acc2, false, false);
                    acc3 = __builtin_amdgcn_wmma_f32_16x16x4_f32(false, a, false, b3,
                                                                 (short)0, acc3, false, false);
                }

                // spill this wave's P band (D layout: VGPR r -> rows r / r+8)
                const int hi8 = (lane >> 4) << 3;
                #pragma unroll
                for (int r = 0; r < 8; ++r) {
                    Plds[wave][r + hi8][ 0 + i16] = acc0[r];
                    Plds[wave][r + hi8][16 + i16] = acc1[r];
                    Plds[wave][r + hi8][32 + i16] = acc2[r];
                    Plds[wave][r + hi8][48 + i16] = acc3[r];
                }
                asm volatile("s_wait_dscnt 0" ::: "memory");  // same-wave LDS RAW

                // gather band: out[.., dxi, h, w0+i] = P[i, i + 2*dxi] / C
                for (int t = lane; t < NDISP * 16; t += 32) {
                    int dxi = t >> 4, i = t & 15;
                    outp[(size_t)dxi * planeHW + i] = Plds[wave][i][i + 2 * dxi] * invC;
                }
            } else {
                for (int t = lane; t < NDISP * 16; t += 32)
                    outp[(size_t)(t >> 4) * planeHW + (t & 15)] = 0.0f;
            }
        }

#ifndef HAVE_ASYNC_LDS
        // commit next slot into the alternate buffer (register pipeline path)
        if (nok) {
            #pragma unroll
            for (int r = 0; r < 16; ++r)
                if (bxok) *(float4*)&Blds[buf ^ 1][bc0 + (r << 4)][bj4] = breg[r];
        }
#else
        asm volatile("s_wait_asynccnt 0" ::: "memory");   // next buffer resident
#endif
        __syncthreads();
    }
}

extern "C" void kernel_launch(void* const* d_in, const int* in_sizes, int n_in,
                              void* d_out, int out_size, void* d_ws, size_t ws_size,
                              hipStream_t stream)
{
    (void)in_sizes; (void)n_in; (void)d_ws; (void)ws_size; (void)out_size;
    const float* in1 = (const float*)d_in[0];
    const float* in2 = (const float*)d_in[1];
    float* out = (float*)d_out;

    dim3 grid(WIDTH / 16, 2 * (HEIGHT / 2 / GH), BATCH);  // (8, 12, 8)
    dim3 block(256);                                       // 8 waves: one h row each
    corr_wmma_kernel<<<grid, block, 0, stream>>>(in1, in2, out);
}